// JK_83726092468482
// MI455X (gfx1250) — compile-verified
//
#include <hip/hip_runtime.h>

typedef float v2f __attribute__((ext_vector_type(2)));
typedef float v8f __attribute__((ext_vector_type(8)));

#define N_NODES   50000
#define N_EDGES   800000
#define F_IN      128
#define L_LAYERS  4
#define LIN_DIM   256
#define OUT_DIM   10
#define N_GRAPHS  64
#define JK_DIM    (L_LAYERS * F_IN)   // 512

// ---------------------------------------------------------------- utilities
__global__ void k_fill(float* __restrict__ p, float v, int n) {
    int i = blockIdx.x * blockDim.x + threadIdx.x;
    if (i < n) p[i] = v;
}

// deg[dst] += 1 for every edge (deg pre-initialized to 1.0 for self-loops)
__global__ void k_deg(const int* __restrict__ dst, float* __restrict__ deg) {
    int e = blockIdx.x * blockDim.x + threadIdx.x;   // grid sized exactly to E
    atomicAdd(&deg[dst[e]], 1.0f);
}

__global__ void k_rsqrt_inplace(float* __restrict__ p, int n) {
    int i = blockIdx.x * blockDim.x + threadIdx.x;
    if (i < n) p[i] = rsqrtf(p[i]);
}

// ------------------------------------------------- fp32 WMMA GEMM: T = H @ W
// Block = 256 threads (8 waves) handles a 16-row strip; wave w owns columns
// [16w, 16w+16). K=128 consumed in 32 steps of V_WMMA_F32_16X16X4_F32.
__global__ void __launch_bounds__(256)
k_gemm_wmma(const float* __restrict__ H, const float* __restrict__ W,
            float* __restrict__ T) {
    __shared__ float As[16 * 132];          // pad 128->132: conflict-free reads
    const int row0 = blockIdx.x * 16;
    const int lane = threadIdx.x & 31;
    const int wave = threadIdx.x >> 5;
    const int n0   = wave * 16;

    // cooperative load of the 16x128 A-tile (2048 floats, 8 per thread)
    #pragma unroll
    for (int i = 0; i < 8; ++i) {
        int idx = threadIdx.x + i * 256;
        int r = idx >> 7, c = idx & 127;
        As[r * 132 + c] = H[(row0 + r) * F_IN + c];
    }
    __syncthreads();

    const int m  = lane & 15;               // M (for A) == N offset (for B)
    const int kh = (lane >> 4) << 1;        // lanes 16-31 hold K+2, K+3

    v8f acc = {};
    #pragma unroll 4
    for (int k = 0; k < F_IN; k += 4) {
        v2f a, b;
        a.x = As[m * 132 + k + kh];
        a.y = As[m * 132 + k + kh + 1];
        b.x = W[(k + kh)     * F_IN + n0 + m];
        b.y = W[(k + kh + 1) * F_IN + n0 + m];
        acc = __builtin_amdgcn_wmma_f32_16x16x4_f32(
                  false, a, false, b, (short)0, acc, false, false);
    }

    const int mh = 8 * (lane >> 4);         // C/D: VGPR r -> M = r (+8 hi half)
    #pragma unroll
    for (int r = 0; r < 8; ++r)
        T[(row0 + r + mh) * F_IN + n0 + m] = acc[r];
}

// ------------------------------------------------------------- aggregation
// self-loop term: Hout = T * dinv^2   (also initializes the accumulator)
__global__ void k_self(const float* __restrict__ T, const float* __restrict__ dinv,
                       float* __restrict__ Hout) {
    int i = blockIdx.x * blockDim.x + threadIdx.x;   // grid == N*F exactly
    int node = i >> 7;
    float dv = dinv[node];
    Hout[i] = T[i] * dv * dv;
}

// one wave per edge: lane handles 4 contiguous features (32*4 = 128)
__global__ void __launch_bounds__(256)
k_scatter(const int* __restrict__ src, const int* __restrict__ dst,
          const float* __restrict__ dinv, const float* __restrict__ T,
          float* __restrict__ Hout) {
    int e    = blockIdx.x * 8 + (threadIdx.x >> 5);  // grid sized exactly
    int lane = threadIdx.x & 31;
    int s = src[e], d = dst[e];
    float nrm = dinv[s] * dinv[d];
    const float4 tv = *(const float4*)(T + s * F_IN + lane * 4);
    float* o = Hout + d * F_IN + lane * 4;
    atomicAdd(o + 0, tv.x * nrm);
    atomicAdd(o + 1, tv.y * nrm);
    atomicAdd(o + 2, tv.z * nrm);
    atomicAdd(o + 3, tv.w * nrm);
}

// bias + ReLU in place, fused JK global-max-pool (values >= 0 -> int atomicMax)
__global__ void k_bias_relu_pool(float* __restrict__ H, const float* __restrict__ b,
                                 const int* __restrict__ batch,
                                 float* __restrict__ g, int layer) {
    int i = blockIdx.x * blockDim.x + threadIdx.x;   // grid == N*F exactly
    int node = i >> 7, f = i & 127;
    float v = H[i] + b[f];
    v = v > 0.0f ? v : 0.0f;
    H[i] = v;
    int gr = batch[node];
    atomicMax((int*)&g[gr * JK_DIM + layer * F_IN + f], __float_as_int(v));
}

// ------------------------------------------------------------------- MLP
__global__ void k_mlp1(const float* __restrict__ g, const float* __restrict__ Wlin,
                       const float* __restrict__ blin, float* __restrict__ hid) {
    int i = blockIdx.x * blockDim.x + threadIdx.x;   // 64*256 exactly
    int gr = i >> 8, j = i & 255;
    float acc = blin[j];
    const float* grow = g + gr * JK_DIM;
    for (int k = 0; k < JK_DIM; ++k)
        acc += grow[k] * Wlin[k * LIN_DIM + j];
    hid[i] = acc > 0.0f ? acc : 0.0f;
}

__global__ void k_mlp2(const float* __restrict__ hid, const float* __restrict__ Wout,
                       const float* __restrict__ bout, float* __restrict__ out) {
    int i = blockIdx.x * blockDim.x + threadIdx.x;
    if (i >= N_GRAPHS * OUT_DIM) return;
    int gr = i / OUT_DIM, o = i % OUT_DIM;
    float acc = bout[o];
    const float* hrow = hid + gr * LIN_DIM;
    for (int j = 0; j < LIN_DIM; ++j)
        acc += hrow[j] * Wout[j * OUT_DIM + o];
    out[i] = acc;
}

// ---------------------------------------------------------------- launcher
extern "C" void kernel_launch(void* const* d_in, const int* in_sizes, int n_in,
                              void* d_out, int out_size, void* d_ws, size_t ws_size,
                              hipStream_t stream) {
    const float* x    = (const float*)d_in[0];
    const int*   ei   = (const int*)  d_in[1];
    const int*   bat  = (const int*)  d_in[2];
    // d_in[3] = pos (unused by the reference result)
    const float* Ws   = (const float*)d_in[4];
    const float* bs   = (const float*)d_in[5];
    const float* Wlin = (const float*)d_in[6];
    const float* blin = (const float*)d_in[7];
    const float* Wout = (const float*)d_in[8];
    const float* bout = (const float*)d_in[9];
    float*       out  = (float*)d_out;

    const int* src = ei;
    const int* dst = ei + N_EDGES;

    // scratch layout (floats): 3 rotating N*F buffers, dinv, pooled g, hidden
    float* bufA = (float*)d_ws;                 // always the GEMM output T
    float* bufB = bufA + (size_t)N_NODES * F_IN;
    float* bufC = bufB + (size_t)N_NODES * F_IN;
    float* dinv = bufC + (size_t)N_NODES * F_IN;
    float* g    = dinv + N_NODES;               // 64 x 512
    float* hid  = g    + N_GRAPHS * JK_DIM;     // 64 x 256

    // ---- degree -> dinv (deg computed in-place in dinv buffer)
    k_fill<<<(N_NODES + 255) / 256, 256, 0, stream>>>(dinv, 1.0f, N_NODES);
    k_deg<<<N_EDGES / 256, 256, 0, stream>>>(dst, dinv);
    k_rsqrt_inplace<<<(N_NODES + 255) / 256, 256, 0, stream>>>(dinv, N_NODES);

    // ---- pooled max accumulator (post-ReLU values >= 0, so 0 is identity)
    k_fill<<<(N_GRAPHS * JK_DIM + 255) / 256, 256, 0, stream>>>(g, 0.0f,
                                                                N_GRAPHS * JK_DIM);

    const int NF_BLOCKS = (N_NODES * F_IN) / 256;   // 25000, exact

    const float* hin = x;
    float* houts[L_LAYERS] = {bufB, bufC, bufB, bufC};
    for (int l = 0; l < L_LAYERS; ++l) {
        float* hout = houts[l];
        k_gemm_wmma<<<N_NODES / 16, 256, 0, stream>>>(hin, Ws + (size_t)l * F_IN * F_IN,
                                                      bufA);
        k_self<<<NF_BLOCKS, 256, 0, stream>>>(bufA, dinv, hout);
        k_scatter<<<N_EDGES / 8, 256, 0, stream>>>(src, dst, dinv, bufA, hout);
        k_bias_relu_pool<<<NF_BLOCKS, 256, 0, stream>>>(hout, bs + l * F_IN,
                                                        bat, g, l);
        hin = hout;
    }

    k_mlp1<<<(N_GRAPHS * LIN_DIM) / 256, 256, 0, stream>>>(g, Wlin, blin, hid);
    k_mlp2<<<(N_GRAPHS * OUT_DIM + 255) / 256, 256, 0, stream>>>(hid, Wout, bout, out);
}